// AttnDecoderRNN_22247930593356
// MI455X (gfx1250) — compile-verified
//
#include <hip/hip_runtime.h>

#define H 256
#define VOCAB 32000
#define BATCH 128
#define TSEQ 2048
#define CH 16                 // t-rows staged per chunk (16 KB)
#define NC (TSEQ / CH)
#define WTS (H + 4)           // padded LDS row stride -> conflict-free b64 frags

typedef __attribute__((ext_vector_type(2))) float v2f;
typedef __attribute__((ext_vector_type(8))) float v8f;
typedef int v4i_g __attribute__((vector_size(16)));   // matches builtin param

// -------- CDNA5 async global->LDS staging (guarded, with sync fallback) ----
#if defined(__HIP_DEVICE_COMPILE__) &&                                   \
    __has_builtin(__builtin_amdgcn_global_load_async_to_lds_b128) &&     \
    __has_builtin(__builtin_amdgcn_s_wait_asynccnt)
#define ASYNC_LDS 1
#endif

__device__ __forceinline__ void cp16(float* dst_lds, const float* src_glb) {
#ifdef ASYNC_LDS
  __builtin_amdgcn_global_load_async_to_lds_b128(
      (__attribute__((address_space(1))) v4i_g*)src_glb,
      (__attribute__((address_space(3))) v4i_g*)dst_lds, 0, 0);
#else
  *(float4*)dst_lds = *(const float4*)src_glb;
#endif
}
template <int N>
__device__ __forceinline__ void async_wait() {
#ifdef ASYNC_LDS
  __builtin_amdgcn_s_wait_asynccnt(N);
#endif
}

// ---------------- block reduction helpers ----------------
__device__ __forceinline__ float block_reduce_sum(float v, float* red) {
  int tid = threadIdx.x;
  red[tid] = v; __syncthreads();
  for (int s = 128; s > 0; s >>= 1) {
    if (tid < s) red[tid] += red[tid + s];
    __syncthreads();
  }
  float r = red[0]; __syncthreads();
  return r;
}
__device__ __forceinline__ float block_reduce_max(float v, float* red) {
  int tid = threadIdx.x;
  red[tid] = v; __syncthreads();
  for (int s = 128; s > 0; s >>= 1) {
    if (tid < s) red[tid] = fmaxf(red[tid], red[tid + s]);
    __syncthreads();
  }
  float r = red[0]; __syncthreads();
  return r;
}

// stage CH rows of enc[t, b, :] into an LDS tile (CH*H floats), 4x b128/thread
__device__ __forceinline__ void stage_chunk(float* dst, const float* enc,
                                            int b, int chunk, int tid) {
#pragma unroll
  for (int i = 0; i < 4; ++i) {
    const int f4 = tid + 256 * i;           // 1024 float4 = 16 KB
    const int row = f4 >> 6, c4 = f4 & 63;  // 64 float4 per 1KB row
    cp16(dst + row * H + c4 * 4,
         enc + ((size_t)(chunk * CH + row) * BATCH + b) * H + c4 * 4);
  }
}

// ------- 1) attention: single pass over enc (online softmax + context) -----
// one workgroup (256 thr = 8 waves) per batch row; enc read exactly once.
__global__ __launch_bounds__(256) void attn_kernel(
    const float* __restrict__ hidden, const float* __restrict__ enc,
    const float* __restrict__ attn_W, const float* __restrict__ attn_b,
    float* __restrict__ out_attnw, float* __restrict__ out_ctx) {
  __shared__ float tile[2][CH * H];   // double-buffered 16 KB chunks
  __shared__ float sc[TSEQ];          // raw scores (8 KB)
  __shared__ float red[256];
  const int b = blockIdx.x;
  const int tid = threadIdx.x;
  const int wave = tid >> 5, lane = tid & 31;

  // s_h = dot(hidden[b], W_h) + attn_b
  float p = hidden[b * H + tid] * attn_W[tid];
  const float s_h = block_reduce_sum(p, red) + attn_b[0];

  float we[8];                        // W_e slice resident in VGPRs
#pragma unroll
  for (int i = 0; i < 8; ++i) we[i] = attn_W[H + lane + 32 * i];

  stage_chunk(tile[0], enc, b, 0, tid);   // prologue prefetch

  float m_run = -1e30f, s_run = 0.f, ctx = 0.f;
  for (int c = 0; c < NC; ++c) {
    const float* tl = tile[c & 1];
    if (c + 1 < NC) {                 // prefetch next chunk, keep 4 in flight
      stage_chunk(tile[(c + 1) & 1], enc, b, c + 1, tid);
      async_wait<4>();
    } else {
      async_wait<0>();
    }
    __syncthreads();                  // tile[c&1] visible to all waves

    // scores: wave w owns rows w and w+8 of the chunk (LDS, conflict-free)
#pragma unroll
    for (int rr = 0; rr < CH / 8; ++rr) {
      const int r = wave + 8 * rr;
      float s = 0.f;
#pragma unroll
      for (int i = 0; i < 8; ++i) s += tl[r * H + lane + 32 * i] * we[i];
      for (int off = 16; off > 0; off >>= 1) s += __shfl_xor(s, off, 32);
      if (lane == 0) sc[c * CH + r] = s + s_h;
    }
    __syncthreads();

    // online softmax update (all threads compute identical m/s from LDS)
    float m_c = sc[c * CH];
#pragma unroll
    for (int r = 1; r < CH; ++r) m_c = fmaxf(m_c, sc[c * CH + r]);
    const float m_new = fmaxf(m_run, m_c);
    const float scl = __expf(m_run - m_new);
    s_run *= scl; ctx *= scl;
#pragma unroll
    for (int r = 0; r < CH; ++r) {
      const float e = __expf(sc[c * CH + r] - m_new);
      s_run += e;
      ctx += e * tl[r * H + tid];     // thread tid owns channel tid
    }
    m_run = m_new;
    __syncthreads();                  // everyone done reading tile[c&1]
  }

  const float inv = 1.f / s_run;
  out_ctx[b * H + tid] = ctx * inv;
#pragma unroll
  for (int i = 0; i < TSEQ / 256; ++i) {
    const int t = tid + 256 * i;
    out_attnw[(size_t)b * TSEQ + t] = __expf(sc[t] - m_run) * inv;
  }
}

// ---------------- 2) embedding + combine(relu) + GRU cell ----------------
__global__ __launch_bounds__(256) void fuse_kernel(
    const int* __restrict__ tokens, const float* __restrict__ hidden,
    const float* __restrict__ emb_table,
    const float* __restrict__ comb_W, const float* __restrict__ comb_b,
    const float* __restrict__ Wih, const float* __restrict__ Whh,
    const float* __restrict__ bih, const float* __restrict__ bhh,
    const float* __restrict__ ctx, float* __restrict__ out_hnew) {
  __shared__ float xin[2 * H];
  __shared__ float xh[H];
  __shared__ float hrow[H];
  const int b = blockIdx.x;
  const int j = threadIdx.x;
  const int tok = tokens[b];
  xin[j]     = emb_table[(size_t)tok * H + j];
  xin[H + j] = ctx[b * H + j];
  hrow[j]    = hidden[b * H + j];
  __syncthreads();

  float acc = comb_b[j];
  const float* crow = comb_W + (size_t)j * (2 * H);
#pragma unroll 8
  for (int k = 0; k < 2 * H; ++k) acc += xin[k] * crow[k];
  xh[j] = fmaxf(acc, 0.f);
  __syncthreads();

  float gxr = bih[j], gxz = bih[H + j], gxn = bih[2 * H + j];
  float ghr = bhh[j], ghz = bhh[H + j], ghn = bhh[2 * H + j];
  const float* wir = Wih + (size_t)j * H;
  const float* wiz = Wih + (size_t)(H + j) * H;
  const float* win = Wih + (size_t)(2 * H + j) * H;
  const float* whr = Whh + (size_t)j * H;
  const float* whz = Whh + (size_t)(H + j) * H;
  const float* whn = Whh + (size_t)(2 * H + j) * H;
#pragma unroll 4
  for (int k = 0; k < H; ++k) {
    const float xv = xh[k], hv = hrow[k];
    gxr += xv * wir[k]; gxz += xv * wiz[k]; gxn += xv * win[k];
    ghr += hv * whr[k]; ghz += hv * whz[k]; ghn += hv * whn[k];
  }
  const float r = 1.f / (1.f + __expf(-(gxr + ghr)));
  const float z = 1.f / (1.f + __expf(-(gxz + ghz)));
  const float n = tanhf(gxn + r * ghn);
  out_hnew[b * H + j] = (1.f - z) * n + z * hrow[j];
}

// ------- 3) logits = h_new @ out_W^T + b via f32 WMMA, LDS-staged B tile ---
// block stages its 16-row out_W tile once (async b128); 8 waves share it.
__global__ __launch_bounds__(256) void logits_wmma_kernel(
    const float* __restrict__ A,      // h_new: BATCH x H
    const float* __restrict__ W,      // out_W: VOCAB x H
    const float* __restrict__ bias,   // out_b
    float* __restrict__ out) {        // raw logits: BATCH x VOCAB
  __shared__ float wt[16 * WTS];      // padded rows -> bank-conflict-free
  const int n0 = blockIdx.x * 16;
  const int tid = threadIdx.x;
#pragma unroll
  for (int i = 0; i < 4; ++i) {
    const int f4 = tid + 256 * i;
    const int row = f4 >> 6, c4 = f4 & 63;
    cp16(wt + row * WTS + c4 * 4, W + (size_t)(n0 + row) * H + c4 * 4);
  }
  async_wait<0>();
  __syncthreads();

  const int wave = tid >> 5;
  const int m0 = wave * 16;           // 8 waves cover M = 128
  const int lane = tid & 31;
  const int half = lane >> 4;         // K sub-pair selector (ISA A/B layout)
  const int l = lane & 15;

  const float* arow = A + (size_t)(m0 + l) * H + 2 * half;   // global, cached
  const float* wrow = wt + l * WTS + 2 * half;               // LDS

  v8f c = {};
#pragma unroll 8
  for (int k = 0; k < H; k += 4) {
    v2f a  = *(const v2f*)(arow + k);
    v2f bv = *(const v2f*)(wrow + k);
    c = __builtin_amdgcn_wmma_f32_16x16x4_f32(
        false, a, false, bv, (short)0, c, false, false);
  }

  const int v = n0 + l;
  const float bb = bias[v];
  const int mbase = m0 + half * 8;    // C/D layout: lanes16-31 hold M=8..15
#pragma unroll
  for (int r = 0; r < 8; ++r)
    out[(size_t)(mbase + r) * VOCAB + v] = c[r] + bb;
}

// ---------------- 4) in-place row log-softmax over VOCAB ----------------
__global__ __launch_bounds__(256) void logsoftmax_kernel(float* __restrict__ logp) {
  __shared__ float red[256];
  const int b = blockIdx.x;
  const int tid = threadIdx.x;
  float* row = logp + (size_t)b * VOCAB;
  float m = -1e30f;
  for (int v = tid; v < VOCAB; v += 256) m = fmaxf(m, row[v]);
  m = block_reduce_max(m, red);
  float s = 0.f;
  for (int v = tid; v < VOCAB; v += 256) s += __expf(row[v] - m);
  s = block_reduce_sum(s, red);
  const float L = m + __logf(s);
  for (int v = tid; v < VOCAB; v += 256) row[v] -= L;
}

extern "C" void kernel_launch(void* const* d_in, const int* in_sizes, int n_in,
                              void* d_out, int out_size, void* d_ws, size_t ws_size,
                              hipStream_t stream) {
  const int*   tokens = (const int*)d_in[0];
  const float* hidden = (const float*)d_in[1];
  const float* enc    = (const float*)d_in[2];
  const float* emb    = (const float*)d_in[3];
  const float* attn_W = (const float*)d_in[4];
  const float* attn_b = (const float*)d_in[5];
  const float* comb_W = (const float*)d_in[6];
  const float* comb_b = (const float*)d_in[7];
  const float* Wih    = (const float*)d_in[8];
  const float* Whh    = (const float*)d_in[9];
  const float* bih    = (const float*)d_in[10];
  const float* bhh    = (const float*)d_in[11];
  const float* out_W  = (const float*)d_in[12];
  const float* out_b  = (const float*)d_in[13];

  float* out   = (float*)d_out;
  float* logp  = out;                               // B*V
  float* hnew  = out + (size_t)BATCH * VOCAB;       // B*H
  float* attnw = hnew + BATCH * H;                  // B*T
  float* ctx   = attnw + (size_t)BATCH * TSEQ;      // B*H

  attn_kernel<<<BATCH, 256, 0, stream>>>(hidden, enc, attn_W, attn_b, attnw, ctx);
  fuse_kernel<<<BATCH, 256, 0, stream>>>(tokens, hidden, emb, comb_W, comb_b,
                                         Wih, Whh, bih, bhh, ctx, hnew);
  logits_wmma_kernel<<<VOCAB / 16, 256, 0, stream>>>(hnew, out_W, out_b, logp);
  logsoftmax_kernel<<<BATCH, 256, 0, stream>>>(logp);
}